// LearnbleFDEINT_47502338294171
// MI455X (gfx1250) — compile-verified
//
#include <hip/hip_runtime.h>
#include <math.h>

typedef float v2f __attribute__((ext_vector_type(2)));
typedef float v4f __attribute__((ext_vector_type(4)));
typedef float v8f __attribute__((ext_vector_type(8)));

#define B_ 64
#define D_ 512
#define S_ 4096
#define H_ 512

// ---------------------------------------------------------------------------
// Kernel A: q[b,h] = sum_d y_cur[b,d] * Wq[h,d] + bq[h]
// WMMA f32 16x16x4, one wave per 16x16 tile of the 64x512 output.
// A fragment: a[m][k] with m=lane&15, k split lanes<16:{k0,k0+1} lanes>=16:{k0+2,k0+3}
// B fragment: b[k][n] with n=lane&15, same k split (B = Wq^T, i.e. b[k=d][n=h]=Wq[h,d])
// D layout: vgpr r, lane l<16 -> (M=r, N=l); l>=16 -> (M=r+8, N=l-16)
// ---------------------------------------------------------------------------
__global__ void qproj_kernel(const float* __restrict__ y,
                             const float* __restrict__ Wq,
                             const float* __restrict__ bq,
                             float* __restrict__ q) {
    int wave = (blockIdx.x * blockDim.x + threadIdx.x) >> 5; // 0..127
    int lane = threadIdx.x & 31;
    int mt = wave >> 5;   // 0..3   (M tiles: 64/16)
    int nt = wave & 31;   // 0..31  (N tiles: 512/16)
    int row = lane & 15;
    int hi  = lane >> 4;

    const float* arow = y  + (size_t)(mt * 16 + row) * D_; // A row (m = row)
    const float* brow = Wq + (size_t)(nt * 16 + row) * D_; // B col (n = row): Wq[h=n][d=k]

    v8f acc = {};
    for (int k0 = 0; k0 < D_; k0 += 4) {
        int k = k0 + 2 * hi;
        v2f a = { arow[k], arow[k + 1] };
        v2f b = { brow[k], brow[k + 1] };
        acc = __builtin_amdgcn_wmma_f32_16x16x4_f32(false, a, false, b,
                                                    (short)0, acc, false, false);
    }

    int n = nt * 16 + row;
    float bias = bq[n];
    int mbase = mt * 16 + 8 * hi;
#pragma unroll
    for (int r = 0; r < 8; ++r)
        q[(size_t)(mbase + r) * H_ + n] = acc[r] + bias;
}

// ---------------------------------------------------------------------------
// Kernel B: v[b,d] = sum_h q[b,h] * Wk[h,d]    (B[k=h][n=d] = Wk[h*D_+d])
// ---------------------------------------------------------------------------
__global__ void vproj_kernel(const float* __restrict__ q,
                             const float* __restrict__ Wk,
                             float* __restrict__ v) {
    int wave = (blockIdx.x * blockDim.x + threadIdx.x) >> 5; // 0..127
    int lane = threadIdx.x & 31;
    int mt = wave >> 5;   // 0..3
    int nt = wave & 31;   // 0..31
    int row = lane & 15;
    int hi  = lane >> 4;

    const float* arow = q + (size_t)(mt * 16 + row) * H_;
    int ncol = nt * 16 + row;

    v8f acc = {};
    for (int k0 = 0; k0 < H_; k0 += 4) {
        int k = k0 + 2 * hi;
        v2f a = { arow[k], arow[k + 1] };
        v2f b = { Wk[(size_t)k * D_ + ncol], Wk[(size_t)(k + 1) * D_ + ncol] };
        acc = __builtin_amdgcn_wmma_f32_16x16x4_f32(false, a, false, b,
                                                    (short)0, acc, false, false);
    }

    int mbase = mt * 16 + 8 * hi;
#pragma unroll
    for (int r = 0; r < 8; ++r)
        v[(size_t)(mbase + r) * D_ + ncol] = acc[r];
}

// ---------------------------------------------------------------------------
// Kernel C: scores[b,s] = (1/sqrt(H)) * sum_d v[b,d] * y_hist[b,d,s]
// The q.bk term cancels in softmax and is omitted.
// Streams 512 MB once -> non-temporal float4 loads, v row staged in LDS.
// grid = (4, 64), block = 256 threads; each thread owns 4 contiguous s.
// ---------------------------------------------------------------------------
__global__ void scores_kernel(const float* __restrict__ v,
                              const float* __restrict__ yh,
                              float* __restrict__ scores) {
    __shared__ float vs[D_];
    int b = blockIdx.y;
    int sbase = blockIdx.x * 1024 + threadIdx.x * 4;

    for (int i = threadIdx.x; i < D_; i += 256)
        vs[i] = v[(size_t)b * D_ + i];
    __syncthreads();

    const float* Yb = yh + (size_t)b * D_ * S_ + sbase;
    v4f acc = {0.f, 0.f, 0.f, 0.f};
#pragma unroll 4
    for (int d = 0; d < D_; ++d) {
        v4f yv = __builtin_nontemporal_load((const v4f*)(Yb + (size_t)d * S_));
        float vv = vs[d];
        acc += vv * yv;
    }
    const float sc = 0.044194173824159216f; // 1/sqrt(512)
    acc *= sc;
    *(v4f*)(scores + (size_t)b * S_ + sbase) = acc;
}

// ---------------------------------------------------------------------------
// Kernel D: per-row softmax over S, then add fractional-power bias.
// In-place on d_out. grid = 64, block = 256.
// ---------------------------------------------------------------------------
__global__ void softmax_bias_kernel(float* __restrict__ out,
                                    const float* __restrict__ beta_p,
                                    const float* __restrict__ h_p) {
    __shared__ float sdata[256];
    int b = blockIdx.x;
    int tid = threadIdx.x;
    float* row = out + (size_t)b * S_;

    float m = -INFINITY;
    for (int s = tid; s < S_; s += 256) m = fmaxf(m, row[s]);
    sdata[tid] = m; __syncthreads();
    for (int off = 128; off > 0; off >>= 1) {
        if (tid < off) sdata[tid] = fmaxf(sdata[tid], sdata[tid + off]);
        __syncthreads();
    }
    float rowmax = sdata[0];
    __syncthreads();

    float sum = 0.f;
    for (int s = tid; s < S_; s += 256) sum += __expf(row[s] - rowmax);
    sdata[tid] = sum; __syncthreads();
    for (int off = 128; off > 0; off >>= 1) {
        if (tid < off) sdata[tid] += sdata[tid + off];
        __syncthreads();
    }
    float inv = 1.0f / sdata[0];

    float be = beta_p[0];
    float hh = h_p[0];
    float coef = powf(hh, be) / be;
    for (int s = tid; s < S_; s += 256) {
        float e = __expf(row[s] - rowmax) * inv;
        float bias = coef * powf((float)(S_ - s), be);
        row[s] = e + bias;
    }
}

// ---------------------------------------------------------------------------
// Inputs (setup_inputs order):
// 0 t_current[1]  1 t_history[S]  2 y_current[B*D]  3 y_history[B*D*S]
// 4 beta[1]       5 h[1]          6 Wq[H*D]         7 bq[H]
// 8 Wk[H*D]       9 bk[H]   (bk unused: constant shift cancels in softmax)
// ---------------------------------------------------------------------------
extern "C" void kernel_launch(void* const* d_in, const int* in_sizes, int n_in,
                              void* d_out, int out_size, void* d_ws, size_t ws_size,
                              hipStream_t stream) {
    const float* y_cur  = (const float*)d_in[2];
    const float* y_hist = (const float*)d_in[3];
    const float* beta   = (const float*)d_in[4];
    const float* h      = (const float*)d_in[5];
    const float* Wq     = (const float*)d_in[6];
    const float* bq     = (const float*)d_in[7];
    const float* Wk     = (const float*)d_in[8];

    float* q = (float*)d_ws;                 // [B,H]  128 KB
    float* v = q + (size_t)B_ * H_;          // [B,D]  128 KB
    float* out = (float*)d_out;              // [B,S]  scores, then final in-place

    // 128 waves -> 16 blocks x 256 threads (8 waves/block)
    qproj_kernel<<<16, 256, 0, stream>>>(y_cur, Wq, bq, q);
    vproj_kernel<<<16, 256, 0, stream>>>(q, Wk, v);

    dim3 gs(S_ / 1024, B_);
    scores_kernel<<<gs, 256, 0, stream>>>(v, y_hist, out);

    softmax_bias_kernel<<<B_, 256, 0, stream>>>(out, beta, h);
}